// FraudGIN_73907797230233
// MI455X (gfx1250) — compile-verified
//
#include <hip/hip_runtime.h>
#include <hip/hip_bf16.h>

typedef __attribute__((ext_vector_type(2))) float v2f;
typedef __attribute__((ext_vector_type(8))) float v8f;

#define TS 132   // padded row stride (floats): 132 % 64 == 4 -> bank-conflict-free columns, float4 aligned
#define CTS 68   // classifier t-tile stride

// ---------------------------------------------------------------- zero agg
__global__ __launch_bounds__(256) void zero_kernel(float4* __restrict__ p, long n4) {
    long i = (long)blockIdx.x * 256 + threadIdx.x;
    if (i < n4) p[i] = make_float4(0.f, 0.f, 0.f, 0.f);
}

// ---------------------------------------------------------------- scatter-sum: agg[dst] += x[src]
// One wave per edge; lane l moves features [4l, 4l+3]. x+agg are L2-resident (102MB < 192MB L2).
__global__ __launch_bounds__(256) void agg_kernel(const float* __restrict__ x,
                                                  const int* __restrict__ src,
                                                  const int* __restrict__ dst,
                                                  float* __restrict__ agg, int E) {
    int t = blockIdx.x * 256 + threadIdx.x;
    int e = t >> 5;
    int lane = t & 31;
    if (e >= E) return;
    int s = src[e];
    int d = dst[e];
    float4 v = ((const float4*)(x + (size_t)s * 128))[lane];
    float* p = agg + (size_t)d * 128 + lane * 4;
    __hip_atomic_fetch_add(p + 0, v.x, __ATOMIC_RELAXED, __HIP_MEMORY_SCOPE_AGENT);
    __hip_atomic_fetch_add(p + 1, v.y, __ATOMIC_RELAXED, __HIP_MEMORY_SCOPE_AGENT);
    __hip_atomic_fetch_add(p + 2, v.z, __ATOMIC_RELAXED, __HIP_MEMORY_SCOPE_AGENT);
    __hip_atomic_fetch_add(p + 3, v.w, __ATOMIC_RELAXED, __HIP_MEMORY_SCOPE_AGENT);
}

// ---------------------------------------------------------------- fused GIN layer
// out = relu(BN( relu(((1+eps)*in + agg) @ W1 + b1) @ W2 + b2 ))
// 4 waves * 16 rows = 64 rows per workgroup. Full-precision f32 WMMA 16x16x4.
// K outer / N-tiles inner: 8 live accumulators (64 VGPRs), one A-frag load feeds
// 8 independent WMMAs -> low register pressure, minimal operand shuffling.
__global__ __launch_bounds__(128) void gin_mlp_kernel(
    const float* __restrict__ in, const float* __restrict__ agg,
    const float* __restrict__ W1, const float* __restrict__ b1,
    const float* __restrict__ W2, const float* __restrict__ b2,
    const float* __restrict__ gam, const float* __restrict__ bet,
    const float* __restrict__ rmean, const float* __restrict__ rvar,
    const float* __restrict__ epsArr, int layer,
    float* __restrict__ out, int N) {
    __shared__ float Ws[128 * 128];       // 64 KB, holds W1 then W2
    __shared__ float At[4][16 * TS];      // per-wave input tile
    __shared__ float Tt[4][16 * TS];      // per-wave intermediate tile

    const int tid = threadIdx.x;
    const int wave = tid >> 5;
    const int lane = tid & 31;
    const int halfHi = (lane < 16) ? 0 : 1;   // which 16-lane half
    const int kb = halfHi ? 2 : 0;            // A/B fragment K sub-offset
    const int mrow = lane & 15;
    const int waveRow = blockIdx.x * 64 + wave * 16;

    const float e1 = 1.0f + epsArr[layer];

    // stage W1 (16384 floats / 128 threads, float4)
    {
        const float4* s = (const float4*)W1;
        float4* d = (float4*)Ws;
        for (int i = tid; i < 128 * 128 / 4; i += 128) d[i] = s[i];
    }
    // stage A tile: (1+eps)*in + agg  (row-clamped for the tail workgroup)
    for (int r = 0; r < 16; ++r) {
        int row = waveRow + r;
        int sr = (row < N) ? row : 0;
        float4 xi = ((const float4*)(in + (size_t)sr * 128))[lane];
        float4 ai = ((const float4*)(agg + (size_t)sr * 128))[lane];
        float4 h;
        h.x = e1 * xi.x + ai.x;
        h.y = e1 * xi.y + ai.y;
        h.z = e1 * xi.z + ai.z;
        h.w = e1 * xi.w + ai.w;
        *(float4*)&At[wave][r * TS + lane * 4] = h;
    }
    __syncthreads();

    // GEMM1: Tt = relu(At @ W1 + b1)
    {
        v8f acc[8];
#pragma unroll
        for (int nt = 0; nt < 8; ++nt) {
            float bv = b1[nt * 16 + mrow];
            v8f t = {bv, bv, bv, bv, bv, bv, bv, bv};
            acc[nt] = t;
        }
        const float* Ap = &At[wave][mrow * TS];
#pragma unroll 2
        for (int k0 = 0; k0 < 128; k0 += 4) {
            v2f a = *(const v2f*)&Ap[k0 + kb];
            const float* Bp = &Ws[(k0 + kb) * 128 + mrow];
#pragma unroll
            for (int nt = 0; nt < 8; ++nt) {
                v2f b;
                b.x = Bp[nt * 16];
                b.y = Bp[128 + nt * 16];
                acc[nt] = __builtin_amdgcn_wmma_f32_16x16x4_f32(false, a, false, b,
                                                                (short)0, acc[nt], false, false);
            }
        }
#pragma unroll
        for (int nt = 0; nt < 8; ++nt)
            for (int r = 0; r < 8; ++r) {
                int m = r + halfHi * 8;
                float v = acc[nt][r];
                Tt[wave][m * TS + nt * 16 + mrow] = v > 0.f ? v : 0.f;
            }
    }
    __syncthreads();
    // stage W2 over W1
    {
        const float4* s = (const float4*)W2;
        float4* d = (float4*)Ws;
        for (int i = tid; i < 128 * 128 / 4; i += 128) d[i] = s[i];
    }
    __syncthreads();

    // GEMM2 + BN(eval) + relu, streamed to out
    {
        v8f acc[8];
#pragma unroll
        for (int nt = 0; nt < 8; ++nt) {
            float bv = b2[nt * 16 + mrow];
            v8f t = {bv, bv, bv, bv, bv, bv, bv, bv};
            acc[nt] = t;
        }
        const float* Ap = &Tt[wave][mrow * TS];
#pragma unroll 2
        for (int k0 = 0; k0 < 128; k0 += 4) {
            v2f a = *(const v2f*)&Ap[k0 + kb];
            const float* Bp = &Ws[(k0 + kb) * 128 + mrow];
#pragma unroll
            for (int nt = 0; nt < 8; ++nt) {
                v2f b;
                b.x = Bp[nt * 16];
                b.y = Bp[128 + nt * 16];
                acc[nt] = __builtin_amdgcn_wmma_f32_16x16x4_f32(false, a, false, b,
                                                                (short)0, acc[nt], false, false);
            }
        }
#pragma unroll
        for (int nt = 0; nt < 8; ++nt) {
            int n = nt * 16 + mrow;
            float g = gam[n], bt = bet[n], mu = rmean[n];
            float iv = rsqrtf(rvar[n] + 1e-5f);
            for (int r = 0; r < 8; ++r) {
                int m = r + halfHi * 8;
                int row = waveRow + m;
                if (row < N) {
                    float z = (acc[nt][r] - mu) * iv * g + bt;
                    out[(size_t)row * 128 + n] = z > 0.f ? z : 0.f;
                }
            }
        }
    }
}

// ---------------------------------------------------------------- classifier
// out = relu(h @ Wc1 + bc1) @ Wc2 + bc2   (128 -> 64 -> 2)
__global__ __launch_bounds__(128) void classifier_kernel(
    const float* __restrict__ h,
    const float* __restrict__ Wc1, const float* __restrict__ bc1,
    const float* __restrict__ Wc2, const float* __restrict__ bc2,
    float* __restrict__ out, int N) {
    __shared__ float Ws[128 * 64];      // 32 KB
    __shared__ float At[4][16 * TS];
    __shared__ float Tt[4][16 * CTS];
    __shared__ float W2s[128];          // 64x2

    const int tid = threadIdx.x;
    const int wave = tid >> 5;
    const int lane = tid & 31;
    const int halfHi = (lane < 16) ? 0 : 1;
    const int kb = halfHi ? 2 : 0;
    const int mrow = lane & 15;
    const int waveRow = blockIdx.x * 64 + wave * 16;

    {
        const float4* s = (const float4*)Wc1;
        float4* d = (float4*)Ws;
        for (int i = tid; i < 128 * 64 / 4; i += 128) d[i] = s[i];
        if (tid < 128) W2s[tid] = Wc2[tid];
    }
    for (int r = 0; r < 16; ++r) {
        int row = waveRow + r;
        int sr = (row < N) ? row : 0;
        float4 xi = ((const float4*)(h + (size_t)sr * 128))[lane];
        *(float4*)&At[wave][r * TS + lane * 4] = xi;
    }
    __syncthreads();

    // GEMM1: Tt = relu(At @ Wc1 + bc1), 4 column tiles of 16
    {
        v8f acc[4];
#pragma unroll
        for (int nt = 0; nt < 4; ++nt) {
            float bv = bc1[nt * 16 + mrow];
            v8f t = {bv, bv, bv, bv, bv, bv, bv, bv};
            acc[nt] = t;
        }
        const float* Ap = &At[wave][mrow * TS];
#pragma unroll 2
        for (int k0 = 0; k0 < 128; k0 += 4) {
            v2f a = *(const v2f*)&Ap[k0 + kb];
            const float* Bp = &Ws[(k0 + kb) * 64 + mrow];
#pragma unroll
            for (int nt = 0; nt < 4; ++nt) {
                v2f b;
                b.x = Bp[nt * 16];
                b.y = Bp[64 + nt * 16];
                acc[nt] = __builtin_amdgcn_wmma_f32_16x16x4_f32(false, a, false, b,
                                                                (short)0, acc[nt], false, false);
            }
        }
#pragma unroll
        for (int nt = 0; nt < 4; ++nt)
            for (int r = 0; r < 8; ++r) {
                int m = r + halfHi * 8;
                float v = acc[nt][r];
                Tt[wave][m * CTS + nt * 16 + mrow] = v > 0.f ? v : 0.f;
            }
    }
    __syncthreads();

    // final 64 -> 2: lane = 2*row + col over 16 rows
    int row = lane >> 1;
    int col = lane & 1;
    int grow = waveRow + row;
    float accv = bc2[col];
    const float* tr = &Tt[wave][row * CTS];
    for (int k = 0; k < 64; ++k) accv += tr[k] * W2s[k * 2 + col];
    if (grow < N) out[(size_t)grow * 2 + col] = accv;
}

// ---------------------------------------------------------------- host
extern "C" void kernel_launch(void* const* d_in, const int* in_sizes, int n_in,
                              void* d_out, int out_size, void* d_ws, size_t ws_size,
                              hipStream_t stream) {
    const float* x     = (const float*)d_in[0];
    const int*   ei    = (const int*)d_in[1];
    const float* eps   = (const float*)d_in[2];
    const float* W1    = (const float*)d_in[3];
    const float* b1    = (const float*)d_in[4];
    const float* W2    = (const float*)d_in[5];
    const float* b2    = (const float*)d_in[6];
    const float* gam   = (const float*)d_in[7];
    const float* bet   = (const float*)d_in[8];
    const float* rmean = (const float*)d_in[9];
    const float* rvar  = (const float*)d_in[10];
    const float* Wc1   = (const float*)d_in[11];
    const float* bc1   = (const float*)d_in[12];
    const float* Wc2   = (const float*)d_in[13];
    const float* bc2   = (const float*)d_in[14];

    const int N = in_sizes[0] / 128;
    const int E = in_sizes[1] / 2;

    float* h   = (float*)d_ws;
    float* agg = h + (size_t)N * 128;

    const long n4 = (long)N * 32;                 // N*128/4 float4s
    const int zeroBlocks = (int)((n4 + 255) / 256);
    const int aggBlocks  = (int)(((long)E * 32 + 255) / 256);
    const int gemmBlocks = (N + 63) / 64;

    const float* cur = x;
    for (int l = 0; l < 3; ++l) {
        zero_kernel<<<zeroBlocks, 256, 0, stream>>>((float4*)agg, n4);
        agg_kernel<<<aggBlocks, 256, 0, stream>>>(cur, ei, ei + E, agg, E);
        gin_mlp_kernel<<<gemmBlocks, 128, 0, stream>>>(
            cur, agg,
            W1 + (size_t)l * 128 * 128, b1 + (size_t)l * 128,
            W2 + (size_t)l * 128 * 128, b2 + (size_t)l * 128,
            gam + (size_t)l * 128, bet + (size_t)l * 128,
            rmean + (size_t)l * 128, rvar + (size_t)l * 128,
            eps, l, h, N);
        cur = h;
    }
    classifier_kernel<<<gemmBlocks, 128, 0, stream>>>(h, Wc1, bc1, Wc2, bc2,
                                                      (float*)d_out, N);
}